// Align_20237885898851
// MI455X (gfx1250) — compile-verified
//
#include <hip/hip_runtime.h>

typedef __attribute__((ext_vector_type(2))) float v2f;
typedef __attribute__((ext_vector_type(8))) float v8f;

// One block per batch (64 blocks), 6 wave32 waves per block, one wave per
// 16-row tile of the 91 coordinate rows.
// Wave computes D(16x16) = A(16x4: pred rows, K-padded) x B(4x16: R rows)
// via v_wmma_f32_16x16x4_f32, stages the 16x3 useful result (+translation)
// in LDS (merged ds_store_2addr_b32), then the block writes the 273
// contiguous floats fully coalesced.
__global__ __launch_bounds__(192) void align_rt_wmma(
    const float* __restrict__ pred,   // [64,91,3]
    const float* __restrict__ rvec,   // [64,3]
    const float* __restrict__ tvec,   // [64,3]
    float* __restrict__ out)          // [64,91,3]
{
    __shared__ float smem[96 * 3];    // 91 rows padded to 96, x 3 cols

    const int j    = blockIdx.x;        // batch
    const int tile = threadIdx.x >> 5;  // 0..5 (wave id)
    const int lane = threadIdx.x & 31;  // 0..31
    const int nh   = lane & 15;         // column (B/D) / row-in-half (A) index
    const bool hi  = lane >= 16;

    // ---- quaternion (1,b,c,d)/|q| -> rotation matrix ----
    // Every R entry is quadratic in the normalized quaternion, so
    // R = Q(q_unnorm) * 1/(1+b^2+c^2+d^2). Use v_rcp_f32 (1 ULP) instead of the
    // ~10-op IEEE div expansion: this chain is the serial critical path.
    const float qb = rvec[j * 3 + 0];
    const float qc = rvec[j * 3 + 1];
    const float qd = rvec[j * 3 + 2];
    const float inv = __builtin_amdgcn_rcpf(1.0f + qb * qb + qc * qc + qd * qd);

    const float R00 = (1.0f + qb * qb - qc * qc - qd * qd) * inv;
    const float R01 = (2.0f * qb * qc - 2.0f * qd) * inv;
    const float R02 = (2.0f * qb * qd + 2.0f * qc) * inv;
    const float R10 = (2.0f * qb * qc + 2.0f * qd) * inv;
    const float R11 = (1.0f - qb * qb + qc * qc - qd * qd) * inv;
    const float R12 = (2.0f * qc * qd - 2.0f * qb) * inv;
    const float R20 = (2.0f * qb * qd - 2.0f * qc) * inv;
    const float R21 = (2.0f * qc * qd + 2.0f * qb) * inv;
    const float R22 = (1.0f - qb * qb - qc * qc + qd * qd) * inv;

    // ---- B fragment: 4x16 (K x N), row striped across lanes.
    //   v0: K=0 (lanes 0-15) / K=2 (lanes 16-31)
    //   v1: K=1 (lanes 0-15) / K=3 = zero (lanes 16-31)
    // Columns n>=3 only feed output columns we never store -> don't-care (finite).
    const float rx0 = hi ? R20 : R00;
    const float rx1 = hi ? R21 : R01;
    const float rx2 = hi ? R22 : R02;
    v2f bf;
    bf.x = (nh == 0) ? rx0 : ((nh == 1) ? rx1 : rx2);
    const float ry = (nh == 0) ? R10 : ((nh == 1) ? R11 : R12);
    bf.y = hi ? 0.0f : ry;   // K=3 row must be zero (pairs with A's zero K=3)

    // translation for this lane's output column (clamped so load is in-bounds)
    const float tn = tvec[j * 3 + (nh < 2 ? nh : 2)];

    // ---- A fragment: 16x4 (M x K), branch-free.
    //   lanes 0-15 : row M=nh, K=0 in v0, K=1 in v1
    //   lanes 16-31: row M=nh, K=2 in v0, K=3=0 in v1
    // Tail tile: clamp row to 90; duplicated rows only produce output rows >=91
    // which are never written back.
    const float* pj = pred + (size_t)j * 91 * 3;
    int row = tile * 16 + nh;
    row = (row < 91) ? row : 90;

    v2f af;
    af.x = pj[row * 3 + (hi ? 2 : 0)];
    const float ay = pj[row * 3 + 1];
    af.y = hi ? 0.0f : ay;

    // ---- WMMA: D = A x B + 0  (EXEC all-1s: no divergent branches above) ----
    v8f acc = {};
    acc = __builtin_amdgcn_wmma_f32_16x16x4_f32(
        /*neg_a=*/false, af, /*neg_b=*/false, bf,
        /*c_mod=*/(short)0, acc, /*reuse_a=*/false, /*reuse_b=*/false);

    // ---- stage useful 16x3 block (+t) into LDS; no bounds checks (LDS padded) ----
    // D layout: VGPR r, lanes 0-15 -> (M=r, N=nh); lanes 16-31 -> (M=r+8, N=nh)
    if (nh < 3) {
        const int mbase = tile * 16 + (hi ? 8 : 0);
        float* s = smem + mbase * 3 + nh;
#pragma unroll
        for (int r = 0; r < 8; ++r) {
            s[r * 3] = acc[r] + tn;
        }
    }
    __syncthreads();

    // ---- coalesced writeback of 91*3 = 273 contiguous floats ----
    float* oj = out + (size_t)j * 91 * 3;
#pragma unroll
    for (int i = threadIdx.x; i < 273; i += 192) {
        oj[i] = smem[i];
    }
}

extern "C" void kernel_launch(void* const* d_in, const int* in_sizes, int n_in,
                              void* d_out, int out_size, void* d_ws, size_t ws_size,
                              hipStream_t stream) {
    (void)in_sizes; (void)n_in; (void)out_size; (void)d_ws; (void)ws_size;
    const float* pred = (const float*)d_in[0];   // [64,91,3]
    const float* rvec = (const float*)d_in[1];   // [64,3]
    const float* tvec = (const float*)d_in[2];   // [64,3]
    float* out = (float*)d_out;                  // [64,91,3]

    dim3 grid(64);
    dim3 block(192);   // 6 wave32 waves: one per 16-row tile of the 91 rows
    align_rt_wmma<<<grid, block, 0, stream>>>(pred, rvec, tvec, out);
}